// Encoder_34067680592104
// MI455X (gfx1250) — compile-verified
//
#include <hip/hip_runtime.h>

typedef __attribute__((ext_vector_type(2))) float v2f;
typedef __attribute__((ext_vector_type(8))) float v8f;

#define GCN_N      100000
#define GCN_E      1600000
#define GCN_INC    128
#define GCN_HID    64
#define GCN_OUT    64

// ---------------- degree / normalization ----------------

__global__ void init_deg_kernel(float* __restrict__ deg, int n) {
    int i = blockIdx.x * blockDim.x + threadIdx.x;
    if (i < n) deg[i] = 1.0f;  // self loop
}

__global__ void edge_deg_kernel(const long long* __restrict__ dst,
                                float* __restrict__ deg, int e) {
    int i = blockIdx.x * blockDim.x + threadIdx.x;
    if (i < e) atomicAdd(&deg[(int)dst[i]], 1.0f);
}

__global__ void rsqrt_kernel(float* __restrict__ deg, int n) {
    int i = blockIdx.x * blockDim.x + threadIdx.x;
    if (i < n) deg[i] = rsqrtf(deg[i]);  // deg -> dis in place
}

// ---------------- WMMA f32 GEMM: H[nrows,64] = op(A[nrows,K]) @ W[K,64] ----
// One wave computes a 16-row x 64-col tile using V_WMMA_F32_16X16X4_F32.
// A layout (16x4 f32):  lanes 0-15 m=lane, v0:K=0 v1:K=1 ; lanes 16-31 K=2,3
// B layout (4x16 f32):  lanes 0-15 n=lane, v0:K=0 v1:K=1 ; lanes 16-31 K=2,3
// C/D layout: VGPR i -> row i (lanes 0-15) / row i+8 (lanes 16-31), col=lane%16

template <int K, bool RELU>
__global__ void gemm_wmma_kernel(const float* __restrict__ A,
                                 const float* __restrict__ W,
                                 float* __restrict__ H, int nrows) {
    const int lane = threadIdx.x & 31;
    const int wid  = (blockIdx.x * blockDim.x + threadIdx.x) >> 5;
    const int r0   = wid * 16;
    if (r0 >= nrows) return;  // wave-uniform: EXEC stays all-1s for WMMA

    const int m  = lane & 15;
    const int hi = lane >> 4;

    v8f acc[4] = {};

    const float* arow = A + (size_t)(r0 + m) * K + 2 * hi;
    for (int k0 = 0; k0 < K; k0 += 4) {
        v2f a = *(const v2f*)(arow + k0);
        if (RELU) {
            a.x = fmaxf(a.x, 0.0f);
            a.y = fmaxf(a.y, 0.0f);
        }
        const float* wk = W + (size_t)(k0 + 2 * hi) * 64 + m;
#pragma unroll
        for (int t = 0; t < 4; ++t) {
            v2f b;
            b.x = wk[t * 16];
            b.y = wk[64 + t * 16];
            acc[t] = __builtin_amdgcn_wmma_f32_16x16x4_f32(
                false, a, false, b, (short)0, acc[t], false, false);
        }
    }

#pragma unroll
    for (int t = 0; t < 4; ++t) {
#pragma unroll
        for (int i = 0; i < 8; ++i) {
            const int row = r0 + i + hi * 8;
            const int col = t * 16 + m;
            H[(size_t)row * 64 + col] = acc[t][i];
        }
    }
}

// ---------------- self-loop message + bias init ----------------
// out[n,c] = h[n,c] * dis[n]^2 + b[c]

__global__ void selfloop_bias_kernel(const float* __restrict__ h,
                                     const float* __restrict__ dis,
                                     const float* __restrict__ b,
                                     float* __restrict__ out, int total) {
    int i = blockIdx.x * blockDim.x + threadIdx.x;
    if (i < total) {
        int node = i >> 6;
        int c    = i & 63;
        float d  = dis[node];
        out[i] = h[i] * d * d + b[c];
    }
}

// ---------------- edge scatter-add ----------------
// one thread per (edge, channel-pair): 32 threads per edge cover 64 channels

__global__ void edge_scatter_kernel(const long long* __restrict__ src,
                                    const long long* __restrict__ dst,
                                    const float* __restrict__ dis,
                                    const float* __restrict__ h,
                                    float* __restrict__ out, int e) {
    long long gid = (long long)blockIdx.x * blockDim.x + threadIdx.x;
    if (gid >= (long long)e * 32) return;
    int eidx = (int)(gid >> 5);
    int p    = (int)(gid & 31);
    int s = (int)src[eidx];
    int d = (int)dst[eidx];
    float w = dis[s] * dis[d];
    v2f hv = *(const v2f*)(h + (size_t)s * 64 + 2 * p);
    float* o = out + (size_t)d * 64 + 2 * p;
    atomicAdd(o,     hv.x * w);
    atomicAdd(o + 1, hv.y * w);
}

// ---------------- launcher ----------------

extern "C" void kernel_launch(void* const* d_in, const int* in_sizes, int n_in,
                              void* d_out, int out_size, void* d_ws, size_t ws_size,
                              hipStream_t stream) {
    const float*     x   = (const float*)d_in[0];
    const long long* ei  = (const long long*)d_in[1];
    const float*     W1  = (const float*)d_in[2];
    const float*     b1  = (const float*)d_in[3];
    const float*     W2  = (const float*)d_in[4];
    const float*     b2  = (const float*)d_in[5];
    float*           out = (float*)d_out;

    const long long* src = ei;
    const long long* dst = ei + GCN_E;

    // workspace layout
    float* dis  = (float*)d_ws;                 // N floats (deg -> dis)
    float* h1   = dis  + 100352;                // N*64
    float* agg1 = h1   + (size_t)GCN_N * 64;    // N*64
    float* h2   = agg1 + (size_t)GCN_N * 64;    // N*64

    const int feat_total = GCN_N * 64;

    // 1) degree + rsqrt
    init_deg_kernel<<<(GCN_N + 255) / 256, 256, 0, stream>>>(dis, GCN_N);
    edge_deg_kernel<<<(GCN_E + 255) / 256, 256, 0, stream>>>(dst, dis, GCN_E);
    rsqrt_kernel<<<(GCN_N + 255) / 256, 256, 0, stream>>>(dis, GCN_N);

    // 2) layer 1 GEMM: h1 = x @ W1   (6250 waves, 8 waves per block)
    {
        int waves  = GCN_N / 16;
        int blocks = (waves + 7) / 8;
        gemm_wmma_kernel<GCN_INC, false><<<blocks, 256, 0, stream>>>(x, W1, h1, GCN_N);
    }

    // 3) layer 1 aggregation
    selfloop_bias_kernel<<<(feat_total + 255) / 256, 256, 0, stream>>>(
        h1, dis, b1, agg1, feat_total);
    {
        long long threads = (long long)GCN_E * 32;
        int blocks = (int)((threads + 255) / 256);
        edge_scatter_kernel<<<blocks, 256, 0, stream>>>(src, dst, dis, h1, agg1, GCN_E);
    }

    // 4) layer 2 GEMM with fused ReLU on input: h2 = relu(agg1) @ W2
    {
        int waves  = GCN_N / 16;
        int blocks = (waves + 7) / 8;
        gemm_wmma_kernel<GCN_HID, true><<<blocks, 256, 0, stream>>>(agg1, W2, h2, GCN_N);
    }

    // 5) layer 2 aggregation -> d_out
    selfloop_bias_kernel<<<(feat_total + 255) / 256, 256, 0, stream>>>(
        h2, dis, b2, out, feat_total);
    {
        long long threads = (long long)GCN_E * 32;
        int blocks = (int)((threads + 255) / 256);
        edge_scatter_kernel<<<blocks, 256, 0, stream>>>(src, dst, dis, h2, out, GCN_E);
    }
}